// ASAPBlock_53120155517526
// MI455X (gfx1250) — compile-verified
//
#include <hip/hip_runtime.h>
#include <math.h>

// Problem constants (match reference)
#define NNODES 4096
#define NEDGES 131072
#define EPRIME (NEDGES + NNODES)   // 135168 edges incl self loops
#define FIN    128
#define HID    256
#define KTOP   2048                // NNODES/2
#define NEG_SLOPE 0.2f

// Output layout (floats, concatenated in reference return order)
#define OFF_XNEW  0
#define OFF_ANEW  (OFF_XNEW + KTOP * HID)              // 524288
#define OFF_PERM  (OFF_ANEW + KTOP * KTOP)             // 4718592
#define OFF_FIT   (OFF_PERM + KTOP)                    // 4720640
#define OFF_SCORE (OFF_FIT + NNODES)                   // 4724736
#define OFF_XEMB  (OFF_SCORE + EPRIME)                 // 4859904

typedef __attribute__((ext_vector_type(2))) float v2f;
typedef __attribute__((ext_vector_type(8))) float v8f;

// ---------------------------------------------------------------- utilities
__device__ __forceinline__ void edge_rc(int e, const int* __restrict__ row,
                                        const int* __restrict__ col,
                                        int& r, int& c) {
  if (e < NEDGES) { r = row[e]; c = col[e]; }
  else            { r = e - NEDGES; c = e - NEDGES; }
}

__device__ __forceinline__ void atomicMaxF(float* addr, float val) {
  int* ia = (int*)addr;
  int old = __float_as_int(*addr);
  while (__int_as_float(old) < val) {
    int assumed = old;
    old = atomicCAS(ia, assumed, __float_as_int(val));
    if (old == assumed) break;
  }
}

__global__ void k_fill(float* p, float v, int n) {
  int i = blockIdx.x * blockDim.x + threadIdx.x;
  if (i < n) p[i] = v;
}

// ---------------------------------------------------------------- degree / norm
__global__ void k_deg_scatter(const int* __restrict__ col, float* deg) {
  int e = blockIdx.x * blockDim.x + threadIdx.x;
  if (e < NEDGES) atomicAdd(&deg[col[e]], 1.0f);
}

__global__ void k_dinv(const float* __restrict__ deg, float* dinv) {
  int i = blockIdx.x * blockDim.x + threadIdx.x;
  if (i < NNODES) {
    float d = deg[i];
    dinv[i] = (d > 0.0f) ? rsqrtf(d) : 0.0f;
  }
}

// ---------------------------------------------------------------- WMMA fp32 GEMM
// C[MxN] = A[MxK] * B[KxN] (+ bias[N]).  Each wave32 computes a 16x64 C strip
// (4 accumulators) so one A fragment feeds 4 V_WMMA_F32_16X16X4_F32 per K-step:
// 4 MACs-per-lane per A-load instead of 1, amortizing vmem+salu 4x.
__global__ __launch_bounds__(256)
void k_gemm_wmma(const float* __restrict__ A, const float* __restrict__ B,
                 const float* __restrict__ bias, float* __restrict__ C,
                 int M, int Nn, int Kd) {
  int wave = (blockIdx.x * (int)blockDim.x + (int)threadIdx.x) >> 5;
  int lane = threadIdx.x & 31;
  int tN4 = Nn >> 6;                       // number of 64-wide strips
  int m0 = (wave / tN4) << 4;
  int n0 = (wave % tN4) << 6;
  if (m0 >= M) return;
  int half = lane >> 4;                    // lanes 16..31 hold K+2 / M+8 halves
  int l16  = lane & 15;
  v8f acc[4] = {{}, {}, {}, {}};
  const float* Arow = A + (size_t)(m0 + l16) * Kd;
  for (int k0 = 0; k0 < Kd; k0 += 4) {
    int ka = k0 + half * 2;
    v2f a;
    a.x = Arow[ka];
    a.y = Arow[ka + 1];
    const float* B0 = B + (size_t)ka * Nn + n0 + l16;
    const float* B1 = B + (size_t)(ka + 1) * Nn + n0 + l16;
#pragma unroll
    for (int t = 0; t < 4; ++t) {
      v2f b;
      b.x = B0[16 * t];
      b.y = B1[16 * t];
      acc[t] = __builtin_amdgcn_wmma_f32_16x16x4_f32(
          false, a, false, b, (short)0, acc[t], false, false);
    }
  }
#pragma unroll
  for (int t = 0; t < 4; ++t) {
    float bv = bias ? bias[n0 + 16 * t + l16] : 0.0f;
#pragma unroll
    for (int v = 0; v < 8; ++v) {
      int m = m0 + half * 8 + v;
      C[(size_t)m * Nn + n0 + 16 * t + l16] = acc[t][v] + bv;
    }
  }
}

// ---------------------------------------------------------------- GCN aggregation
// g[col,:] += h[row,:] * dinv[row]*dinv[col]   (one block per edge, float4)
__global__ __launch_bounds__(64)
void k_gcn_scatter(const float* __restrict__ h, const int* __restrict__ row,
                   const int* __restrict__ col, const float* __restrict__ dinv,
                   float* __restrict__ g) {
  int e = blockIdx.x;
  int r = row[e], c = col[e];
  float w = dinv[r] * dinv[c];
  const float4* hr = (const float4*)(h + (size_t)r * HID);
  float* gc = g + (size_t)c * HID;
  int t = threadIdx.x;                 // 0..63 -> 64 float4 = 256 floats
  float4 v = hr[t];
  atomicAdd(&gc[4 * t + 0], v.x * w);
  atomicAdd(&gc[4 * t + 1], v.y * w);
  atomicAdd(&gc[4 * t + 2], v.z * w);
  atomicAdd(&gc[4 * t + 3], v.w * w);
}

// dst = relu(g + h*dinv^2 (self loop) + b)
__global__ void k_gcn_epi(float* __restrict__ dst, const float* __restrict__ g,
                          const float* __restrict__ h, const float* __restrict__ dinv,
                          const float* __restrict__ b) {
  int idx = blockIdx.x * blockDim.x + threadIdx.x;
  if (idx < NNODES * HID) {
    int i = idx / HID, j = idx % HID;
    float d = dinv[i];
    float v = g[idx] + h[idx] * d * d + b[j];
    dst[idx] = v > 0.0f ? v : 0.0f;
  }
}

// ---------------------------------------------------------------- segment max (features)
// xq pre-initialized with x_emb (self loop).  x_emb >= 0 post-relu, so int-ordered
// atomicMax is valid.
__global__ __launch_bounds__(256)
void k_segmax_feat(const float* __restrict__ xemb, const int* __restrict__ row,
                   const int* __restrict__ col, float* __restrict__ xq) {
  int e = blockIdx.x;
  int r = row[e], c = col[e];
  int j = threadIdx.x;
  atomicMax((int*)&xq[(size_t)c * HID + j],
            __float_as_int(xemb[(size_t)r * HID + j]));
}

// ---------------------------------------------------------------- per-node dots
// t1[n] = <xqlin[n,:], attW[0:H]>,  t2[n] = <xemb[n,:], attW[H:2H]>
__global__ __launch_bounds__(256)
void k_att_dots(const float* __restrict__ xqlin, const float* __restrict__ xemb,
                const float* __restrict__ attW, float* __restrict__ t1,
                float* __restrict__ t2) {
  int wave = (blockIdx.x * (int)blockDim.x + (int)threadIdx.x) >> 5;
  int lane = threadIdx.x & 31;
  if (wave >= NNODES) return;
  float s1 = 0.0f, s2 = 0.0f;
  for (int j = lane; j < HID; j += 32) {
    s1 += xqlin[(size_t)wave * HID + j] * attW[j];
    s2 += xemb[(size_t)wave * HID + j] * attW[HID + j];
  }
  for (int off = 16; off > 0; off >>= 1) {
    s1 += __shfl_xor(s1, off);
    s2 += __shfl_xor(s2, off);
  }
  if (lane == 0) { t1[wave] = s1; t2[wave] = s2; }
}

__global__ void k_edge_score(const float* __restrict__ t1, const float* __restrict__ t2,
                             const int* __restrict__ row, const int* __restrict__ col,
                             const float* __restrict__ attb, float* __restrict__ score) {
  int e = blockIdx.x * blockDim.x + threadIdx.x;
  if (e >= EPRIME) return;
  int r, c; edge_rc(e, row, col, r, c);
  float s = t1[c] + t2[r] + attb[0];
  score[e] = s > 0.0f ? s : NEG_SLOPE * s;
}

// ---------------------------------------------------------------- segment softmax
__global__ void k_seg_max_scalar(const float* __restrict__ s, const int* __restrict__ row,
                                 const int* __restrict__ col, float* __restrict__ m) {
  int e = blockIdx.x * blockDim.x + threadIdx.x;
  if (e >= EPRIME) return;
  int r, c; edge_rc(e, row, col, r, c);
  atomicMaxF(&m[c], s[e]);
}

__global__ void k_seg_exp(float* __restrict__ s, const int* __restrict__ row,
                          const int* __restrict__ col, const float* __restrict__ m,
                          float* __restrict__ d) {
  int e = blockIdx.x * blockDim.x + threadIdx.x;
  if (e >= EPRIME) return;
  int r, c; edge_rc(e, row, col, r, c);
  float v = __expf(s[e] - m[c]);
  s[e] = v;
  atomicAdd(&d[c], v);
}

__global__ void k_seg_norm(float* __restrict__ s, const int* __restrict__ row,
                           const int* __restrict__ col, const float* __restrict__ d) {
  int e = blockIdx.x * blockDim.x + threadIdx.x;
  if (e >= EPRIME) return;
  int r, c; edge_rc(e, row, col, r, c);
  s[e] /= d[c];
}

// ---------------------------------------------------------------- weighted aggregate
__global__ __launch_bounds__(64)
void k_agg_scatter(const float* __restrict__ xemb, const int* __restrict__ row,
                   const int* __restrict__ col, const float* __restrict__ score,
                   float* __restrict__ xagg) {
  int e = blockIdx.x;
  int r, c; edge_rc(e, row, col, r, c);
  float w = score[e];
  const float4* src = (const float4*)(xemb + (size_t)r * HID);
  float* dst = xagg + (size_t)c * HID;
  int t = threadIdx.x;
  float4 v = src[t];
  atomicAdd(&dst[4 * t + 0], v.x * w);
  atomicAdd(&dst[4 * t + 1], v.y * w);
  atomicAdd(&dst[4 * t + 2], v.z * w);
  atomicAdd(&dst[4 * t + 3], v.w * w);
}

// ---------------------------------------------------------------- LEConv fitness
__global__ __launch_bounds__(256)
void k_le_dots(const float* __restrict__ xagg, const float* __restrict__ w1,
               const float* __restrict__ b1, const float* __restrict__ w2,
               const float* __restrict__ w3, float* __restrict__ a,
               float* __restrict__ bb, float* __restrict__ c3) {
  int wave = (blockIdx.x * (int)blockDim.x + (int)threadIdx.x) >> 5;
  int lane = threadIdx.x & 31;
  if (wave >= NNODES) return;
  float s1 = 0.0f, s2 = 0.0f, s3 = 0.0f;
  for (int j = lane; j < HID; j += 32) {
    float v = xagg[(size_t)wave * HID + j];
    s1 += v * w1[j];
    s2 += v * w2[j];
    s3 += v * w3[j];
  }
  for (int off = 16; off > 0; off >>= 1) {
    s1 += __shfl_xor(s1, off);
    s2 += __shfl_xor(s2, off);
    s3 += __shfl_xor(s3, off);
  }
  if (lane == 0) { a[wave] = s1 + b1[0]; bb[wave] = s2; c3[wave] = s3; }
}

__global__ void k_sumbb_scatter(const float* __restrict__ bb, const int* __restrict__ row,
                                const int* __restrict__ col, float* __restrict__ sumbb) {
  int e = blockIdx.x * blockDim.x + threadIdx.x;
  if (e >= EPRIME) return;
  int r, c; edge_rc(e, row, col, r, c);
  atomicAdd(&sumbb[c], bb[r]);
}

__global__ void k_fitness(const float* __restrict__ deg, const float* __restrict__ a,
                          const float* __restrict__ sumbb, const float* __restrict__ c3,
                          const float* __restrict__ le3b, float* __restrict__ fit) {
  int n = blockIdx.x * blockDim.x + threadIdx.x;
  if (n < NNODES) {
    float v = deg[n] * a[n] - sumbb[n] + c3[n] + le3b[0];
    fit[n] = 1.0f / (1.0f + __expf(-v));
  }
}

// ---------------------------------------------------------------- top-k (bitonic sort)
__global__ __launch_bounds__(1024)
void k_topk(const float* __restrict__ fit, float* __restrict__ perm_f,
            float* __restrict__ fit_top, int* __restrict__ perm_i,
            int* __restrict__ inv_perm) {
  __shared__ float v[NNODES];
  __shared__ int id[NNODES];
  int tid = threadIdx.x;
  for (int i = tid; i < NNODES; i += 1024) { v[i] = fit[i]; id[i] = i; }
  __syncthreads();
  for (int k = 2; k <= NNODES; k <<= 1) {
    for (int j = k >> 1; j > 0; j >>= 1) {
      for (int i = tid; i < NNODES; i += 1024) {
        int l = i ^ j;
        if (l > i) {
          bool desc = ((i & k) == 0);
          float vi = v[i], vl = v[l];
          int ii = id[i], il = id[l];
          bool l_gt = (vl > vi) || (vl == vi && il < ii);  // (val desc, idx asc)
          bool i_gt = (vi > vl) || (vi == vl && ii < il);
          bool sw = desc ? l_gt : i_gt;
          if (sw) { v[i] = vl; v[l] = vi; id[i] = il; id[l] = ii; }
        }
      }
      __syncthreads();
    }
  }
  for (int i = tid; i < KTOP; i += 1024) {
    perm_i[i]  = id[i];
    perm_f[i]  = (float)id[i];
    fit_top[i] = v[i];
    inv_perm[id[i]] = i;   // inv_perm pre-filled with -1
  }
}

__global__ void k_xnew(const float* __restrict__ xagg, const int* __restrict__ perm_i,
                       const float* __restrict__ fit_top, float* __restrict__ xnew) {
  int idx = blockIdx.x * blockDim.x + threadIdx.x;
  if (idx < KTOP * HID) {
    int k = idx / HID, j = idx % HID;
    xnew[idx] = xagg[(size_t)perm_i[k] * HID + j] * fit_top[k];
  }
}

// ---------------------------------------------------------------- coarsening
// S[r, inv_perm[c]] += score  (dense N x K, L2-resident)
__global__ void k_build_S(const float* __restrict__ score, const int* __restrict__ row,
                          const int* __restrict__ col, const int* __restrict__ inv_perm,
                          float* __restrict__ S) {
  int e = blockIdx.x * blockDim.x + threadIdx.x;
  if (e >= EPRIME) return;
  int r, c; edge_rc(e, row, col, r, c);
  int k1 = inv_perm[c];
  if (k1 >= 0) atomicAdd(&S[(size_t)r * KTOP + k1], score[e]);
}

// AS[r,:] += S[c,:]  per edge (A is 0/1-weighted edge list incl self loops)
__global__ __launch_bounds__(256)
void k_AS(const float* __restrict__ S, const int* __restrict__ row,
          const int* __restrict__ col, float* __restrict__ AS) {
  int e = blockIdx.x;
  int r, c; edge_rc(e, row, col, r, c);
  const float4* src = (const float4*)(S + (size_t)c * KTOP);
  float* dst = AS + (size_t)r * KTOP;
  for (int q = threadIdx.x; q < KTOP / 4; q += blockDim.x) {
    float4 v = src[q];
    atomicAdd(&dst[4 * q + 0], v.x);
    atomicAdd(&dst[4 * q + 1], v.y);
    atomicAdd(&dst[4 * q + 2], v.z);
    atomicAdd(&dst[4 * q + 3], v.w);
  }
}

// A_new[inv_perm[c],:] += score_e * AS[r,:]
__global__ __launch_bounds__(256)
void k_Anew(const float* __restrict__ AS, const float* __restrict__ score,
            const int* __restrict__ row, const int* __restrict__ col,
            const int* __restrict__ inv_perm, float* __restrict__ Anew) {
  int e = blockIdx.x;
  int r, c; edge_rc(e, row, col, r, c);
  int k1 = inv_perm[c];
  if (k1 < 0) return;
  float s = score[e];
  const float4* src = (const float4*)(AS + (size_t)r * KTOP);
  float* dst = Anew + (size_t)k1 * KTOP;
  for (int q = threadIdx.x; q < KTOP / 4; q += blockDim.x) {
    float4 v = src[q];
    atomicAdd(&dst[4 * q + 0], v.x * s);
    atomicAdd(&dst[4 * q + 1], v.y * s);
    atomicAdd(&dst[4 * q + 2], v.z * s);
    atomicAdd(&dst[4 * q + 3], v.w * s);
  }
}

__global__ void k_zero_diag(float* __restrict__ Anew) {
  int k = blockIdx.x * blockDim.x + threadIdx.x;
  if (k < KTOP) Anew[(size_t)k * KTOP + k] = 0.0f;
}

// ---------------------------------------------------------------- launch
extern "C" void kernel_launch(void* const* d_in, const int* in_sizes, int n_in,
                              void* d_out, int out_size, void* d_ws, size_t ws_size,
                              hipStream_t stream) {
  const float* x    = (const float*)d_in[0];
  const int*   ei   = (const int*)d_in[1];
  const float* W1   = (const float*)d_in[2];
  const float* b1   = (const float*)d_in[3];
  const float* W2   = (const float*)d_in[4];
  const float* b2   = (const float*)d_in[5];
  const float* linW = (const float*)d_in[6];
  const float* linb = (const float*)d_in[7];
  const float* attW = (const float*)d_in[8];
  const float* attb = (const float*)d_in[9];
  const float* le1W = (const float*)d_in[10];
  const float* le1b = (const float*)d_in[11];
  const float* le2W = (const float*)d_in[12];
  const float* le3W = (const float*)d_in[13];
  const float* le3b = (const float*)d_in[14];
  const int* row = ei;
  const int* col = ei + NEDGES;

  float* out = (float*)d_out;
  float* out_xnew  = out + OFF_XNEW;
  float* out_Anew  = out + OFF_ANEW;
  float* out_perm  = out + OFF_PERM;
  float* out_fit   = out + OFF_FIT;
  float* out_score = out + OFF_SCORE;
  float* out_xemb  = out + OFF_XEMB;

  // workspace carve (256B aligned)
  char* p = (char*)d_ws;
  auto carve = [&](size_t bytes) -> void* {
    void* r = (void*)p;
    p += (bytes + 255) & ~(size_t)255;
    return r;
  };
  float* deg    = (float*)carve(NNODES * 4);
  float* dinv   = (float*)carve(NNODES * 4);
  float* h1     = (float*)carve((size_t)NNODES * HID * 4);
  float* x1     = (float*)carve((size_t)NNODES * HID * 4);
  float* h2     = (float*)carve((size_t)NNODES * HID * 4);
  float* g2     = (float*)carve((size_t)NNODES * HID * 4);
  float* xq     = (float*)carve((size_t)NNODES * HID * 4);
  float* xqlin  = (float*)carve((size_t)NNODES * HID * 4);
  float* xagg   = (float*)carve((size_t)NNODES * HID * 4);
  float* t1     = (float*)carve(NNODES * 4);
  float* t2     = (float*)carve(NNODES * 4);
  float* segmax = (float*)carve(NNODES * 4);
  float* segsum = (float*)carve(NNODES * 4);
  float* aN     = (float*)carve(NNODES * 4);
  float* bbN    = (float*)carve(NNODES * 4);
  float* c3N    = (float*)carve(NNODES * 4);
  float* sumbb  = (float*)carve(NNODES * 4);
  float* fittop = (float*)carve(KTOP * 4);
  int*   perm_i = (int*)carve(KTOP * 4);
  int*   invp   = (int*)carve(NNODES * 4);
  float* Smat   = (float*)carve((size_t)NNODES * KTOP * 4);
  float* ASmat  = (float*)carve((size_t)NNODES * KTOP * 4);

  const int T = 256;
  const int gN   = (NNODES + T - 1) / T;
  const int gE   = (NEDGES + T - 1) / T;
  const int gEP  = (EPRIME + T - 1) / T;
  const int gNH  = (NNODES * HID + T - 1) / T;
  const int gKH  = (KTOP * HID + T - 1) / T;

  // ---- degree & symmetric norm (self loops -> init deg = 1)
  k_fill<<<gN, T, 0, stream>>>(deg, 1.0f, NNODES);
  k_deg_scatter<<<gE, T, 0, stream>>>(col, deg);
  k_dinv<<<gN, T, 0, stream>>>(deg, dinv);

  // ---- GCN layer 1 (WMMA GEMM + sparse aggregate)
  // waves = (M/16) * (N/64) strips, 8 waves / block
  int waves = (NNODES / 16) * (HID / 64);
  k_gemm_wmma<<<waves / 8, T, 0, stream>>>(x, W1, nullptr, h1, NNODES, HID, FIN);
  hipMemsetAsync(x1, 0, (size_t)NNODES * HID * 4, stream);
  k_gcn_scatter<<<NEDGES, 64, 0, stream>>>(h1, row, col, dinv, x1);
  k_gcn_epi<<<gNH, T, 0, stream>>>(x1, x1, h1, dinv, b1);

  // ---- GCN layer 2 -> x_emb (written directly to output slot)
  k_gemm_wmma<<<waves / 8, T, 0, stream>>>(x1, W2, nullptr, h2, NNODES, HID, HID);
  hipMemsetAsync(g2, 0, (size_t)NNODES * HID * 4, stream);
  k_gcn_scatter<<<NEDGES, 64, 0, stream>>>(h2, row, col, dinv, g2);
  k_gcn_epi<<<gNH, T, 0, stream>>>(out_xemb, g2, h2, dinv, b2);

  // ---- ASAP: master-node max pool then lin (WMMA)
  hipMemcpyAsync(xq, out_xemb, (size_t)NNODES * HID * 4,
                 hipMemcpyDeviceToDevice, stream);              // self loops
  k_segmax_feat<<<NEDGES, 256, 0, stream>>>(out_xemb, row, col, xq);
  k_gemm_wmma<<<waves / 8, T, 0, stream>>>(xq, linW, linb, xqlin, NNODES, HID, HID);

  // ---- attention scores (factorized per-node dots) + segment softmax
  k_att_dots<<<(NNODES * 32) / T, T, 0, stream>>>(xqlin, out_xemb, attW, t1, t2);
  k_edge_score<<<gEP, T, 0, stream>>>(t1, t2, row, col, attb, out_score);
  k_fill<<<gN, T, 0, stream>>>(segmax, -3.4e38f, NNODES);
  k_seg_max_scalar<<<gEP, T, 0, stream>>>(out_score, row, col, segmax);
  hipMemsetAsync(segsum, 0, NNODES * 4, stream);
  k_seg_exp<<<gEP, T, 0, stream>>>(out_score, row, col, segmax, segsum);
  k_seg_norm<<<gEP, T, 0, stream>>>(out_score, row, col, segsum);

  // ---- weighted aggregation x_agg
  hipMemsetAsync(xagg, 0, (size_t)NNODES * HID * 4, stream);
  k_agg_scatter<<<EPRIME, 64, 0, stream>>>(out_xemb, row, col, out_score, xagg);

  // ---- LEConv fitness + sigmoid
  k_le_dots<<<(NNODES * 32) / T, T, 0, stream>>>(xagg, le1W, le1b, le2W, le3W,
                                                 aN, bbN, c3N);
  hipMemsetAsync(sumbb, 0, NNODES * 4, stream);
  k_sumbb_scatter<<<gEP, T, 0, stream>>>(bbN, row, col, sumbb);
  k_fitness<<<gN, T, 0, stream>>>(deg, aN, sumbb, c3N, le3b, out_fit);

  // ---- top-k (bitonic in LDS) + x_new
  hipMemsetAsync(invp, 0xFF, NNODES * 4, stream);   // -1
  k_topk<<<1, 1024, 0, stream>>>(out_fit, out_perm, fittop, perm_i, invp);
  k_xnew<<<gKH, T, 0, stream>>>(xagg, perm_i, fittop, out_xnew);

  // ---- coarsening: A_new = S^T (A S), sparse edge-list form (L2-resident)
  hipMemsetAsync(Smat, 0, (size_t)NNODES * KTOP * 4, stream);
  k_build_S<<<gEP, T, 0, stream>>>(out_score, row, col, invp, Smat);
  hipMemsetAsync(ASmat, 0, (size_t)NNODES * KTOP * 4, stream);
  k_AS<<<EPRIME, 256, 0, stream>>>(Smat, row, col, ASmat);
  hipMemsetAsync(out_Anew, 0, (size_t)KTOP * KTOP * 4, stream);
  k_Anew<<<EPRIME, 256, 0, stream>>>(ASmat, out_score, row, col, invp, out_Anew);
  k_zero_diag<<<(KTOP + T - 1) / T, T, 0, stream>>>(out_Anew);

  (void)in_sizes; (void)n_in; (void)out_size; (void)ws_size;
}